// RelPosEmb_17935783428243
// MI455X (gfx1250) — compile-verified
//
#include <hip/hip_runtime.h>

typedef float v2f __attribute__((ext_vector_type(2)));
typedef float v4f __attribute__((ext_vector_type(4)));
typedef float v8f __attribute__((ext_vector_type(8)));

#define BATCH 2
#define HEADS 8
#define NTOK  2048   // 8*16*16
#define DHEAD 64
#define RW 31
#define RH 31
#define RD 15
#define RTOT 77      // RW+RH+RD
#define RPAD 80
#define TILE 16      // rows per block (= X, so x == row)

__global__ __launch_bounds__(256)
void relpos3d_kernel(const float* __restrict__ q,
                     const float* __restrict__ rel_d,
                     const float* __restrict__ rel_h,
                     const float* __restrict__ rel_w,
                     float* __restrict__ out)
{
    __shared__ float Lsh[TILE * RPAD];   // 16 x 80 logits, 5 KB

    const int bh   = blockIdx.x >> 7;    // / 128 tiles
    const int tile = blockIdx.x & 127;
    const int i0   = tile << 4;          // first token row of tile
    const int ycur = (i0 >> 4) & 15;
    const int zcur = i0 >> 8;

    const int tid  = threadIdx.x;
    const int wave = tid >> 5;
    const int lane = tid & 31;

    // ---------------- Phase 1: L[row, r] = q[i0+row,:] . rel[r,:] ----------
    // via V_WMMA_F32_16X16X4_F32, exact fp32 accumulation.
    if (wave < 5) {                       // 5 N-tiles cover 80 (77 used) cols
        const int mn = lane & 15;         // M for A, N for B/C/D
        const int kh = lane >> 4;         // K-pair select per ISA layout
        const int r  = (wave << 4) + mn;  // global rel column 0..79

        const float* qrow = q + ((size_t)bh * NTOK + i0 + mn) * DHEAD;
        const float* rrow;
        bool valid = true;
        if (r < RW)           rrow = rel_w + (size_t)r * DHEAD;
        else if (r < RW + RH) rrow = rel_h + (size_t)(r - RW) * DHEAD;
        else if (r < RTOT)    rrow = rel_d + (size_t)(r - RW - RH) * DHEAD;
        else { rrow = rel_w; valid = false; }   // zero-pad columns 77..79

        v8f acc = {0.f, 0.f, 0.f, 0.f, 0.f, 0.f, 0.f, 0.f};
        #pragma unroll
        for (int k0 = 0; k0 < DHEAD; k0 += 4) {
            const int k = k0 + (kh << 1);       // lane's K pair: k, k+1
            v2f a = *(const v2f*)(qrow + k);    // A[m, k..k+1]
            v2f b = *(const v2f*)(rrow + k);    // B[k..k+1, n] = rel[r, k..k+1]
            if (!valid) { b.x = 0.f; b.y = 0.f; }
            acc = __builtin_amdgcn_wmma_f32_16x16x4_f32(
                false, a, false, b, (short)0, acc, false, false);
        }
        // C/D layout: VGPR g -> M = g + 8*(lane>=16), N = lane&15
        const int mbase = kh << 3;
        #pragma unroll
        for (int g = 0; g < 8; ++g)
            Lsh[(mbase + g) * RPAD + (wave << 4) + mn] = acc[g];
    }
    __syncthreads();

    // ---------------- Phase 2: stream broadcast-sum to global ---------------
    // out[i0+row, (z1,y1,x1)] = L[row, x1-row+15] + L[row, 31 + y1-y+15]
    //                         + L[row, 62 + z1-z+7]
    float* outp = out + ((size_t)bh * NTOK + i0) * NTOK;
    #pragma unroll 4
    for (int it = 0; it < 32; ++it) {
        const int idx = tid + (it << 8);   // 0..8191 float4 slots (16 x 512)
        const int row = idx >> 9;
        const int c4  = idx & 511;
        const int j   = c4 << 2;           // output column, multiple of 4
        const int x1  = j & 15;
        const int y1  = (j >> 4) & 15;
        const int z1  = j >> 8;

        const float* Lr = Lsh + row * RPAD;
        const float hd  = Lr[RW + 15 + y1 - ycur]
                        + Lr[RW + RH + 7 + z1 - zcur];
        const int wb = 15 - row + x1;      // in [0,30], unaligned -> scalar ds
        v4f v;
        v.x = Lr[wb + 0] + hd;
        v.y = Lr[wb + 1] + hd;
        v.z = Lr[wb + 2] + hd;
        v.w = Lr[wb + 3] + hd;
        __builtin_nontemporal_store(v, (v4f*)(outp + (size_t)row * NTOK + j));
    }
}

extern "C" void kernel_launch(void* const* d_in, const int* in_sizes, int n_in,
                              void* d_out, int out_size, void* d_ws, size_t ws_size,
                              hipStream_t stream)
{
    const float* q     = (const float*)d_in[0];
    const float* rel_d = (const float*)d_in[1];   // (15, 64)
    const float* rel_h = (const float*)d_in[2];   // (31, 64)
    const float* rel_w = (const float*)d_in[3];   // (31, 64)
    float* out = (float*)d_out;                   // (2, 8, 2048, 2048)

    dim3 grid(BATCH * HEADS * (NTOK / TILE));     // 2048 blocks
    relpos3d_kernel<<<grid, 256, 0, stream>>>(q, rel_d, rel_h, rel_w, out);
}